// DualBranchGNN_31301721653660
// MI455X (gfx1250) — compile-verified
//
#include <hip/hip_runtime.h>

typedef __attribute__((ext_vector_type(2))) float v2f;
typedef __attribute__((ext_vector_type(8))) float v8f;

#define TPB 256

// ---------------------------------------------------------------------------
// Generic fp32 WMMA GEMM: C[M,N] = act(A[M,K] @ W[K,N] + bias)
// One wave32 computes one 16x16 output tile using V_WMMA_F32_16X16X4_F32.
// Interior tiles: branch-free unconditional loads (wave-uniform predicate,
// EXEC all-ones for WMMA). Edge tiles / K-tail: clamped addresses + cndmask.
// ---------------------------------------------------------------------------
__global__ void wmma_gemm_kernel(const float* __restrict__ A, int lda,
                                 const float* __restrict__ W, int ldw,
                                 const float* __restrict__ bias,
                                 float* __restrict__ C, int ldc,
                                 int M, int N, int K, int relu)
{
    int ntN   = (N + 15) >> 4;
    int ntM   = (M + 15) >> 4;
    int tiles = ntN * ntM;
    int wave  = blockIdx.x * (blockDim.x >> 5) + (threadIdx.x >> 5);
    if (wave >= tiles) return;               // wave-uniform: EXEC stays all-ones
    int tn   = wave % ntN;
    int tm   = wave / ntN;
    int lane = threadIdx.x & 31;
    int half = lane >> 4;                    // 0: K+0/+1, 1: K+2/+3
    int q    = lane & 15;
    int rowA = tm * 16 + q;
    int colB = tn * 16 + q;

    v8f acc = {};
    int K4 = K & ~3;
    bool interior = (tm * 16 + 16 <= M) && (tn * 16 + 16 <= N) && ((lda & 1) == 0);

    if (interior) {
        const float* ap = A + (size_t)rowA * lda + 2 * half;
        const float* bp = W + (size_t)(2 * half) * ldw + colB;
        int k = 0;
        for (; k + 8 <= K4; k += 8) {
            v2f a0 = *(const v2f*)ap;
            v2f a1 = *(const v2f*)(ap + 4);
            v2f b0, b1;
            b0.x = bp[0];
            b0.y = bp[(size_t)ldw];
            b1.x = bp[(size_t)4 * ldw];
            b1.y = bp[(size_t)5 * ldw];
            __builtin_prefetch(ap + 32, 0, 1);   // -> global_prefetch_b8
            acc = __builtin_amdgcn_wmma_f32_16x16x4_f32(false, a0, false, b0,
                                                        (short)0, acc, false, false);
            acc = __builtin_amdgcn_wmma_f32_16x16x4_f32(false, a1, false, b1,
                                                        (short)0, acc, false, false);
            ap += 8;
            bp += (size_t)8 * ldw;
        }
        for (; k < K4; k += 4) {
            v2f a0 = *(const v2f*)ap;
            v2f b0;
            b0.x = bp[0];
            b0.y = bp[(size_t)ldw];
            acc = __builtin_amdgcn_wmma_f32_16x16x4_f32(false, a0, false, b0,
                                                        (short)0, acc, false, false);
            ap += 4;
            bp += (size_t)4 * ldw;
        }
    }

    {
        int  kstart = interior ? K4 : 0;
        bool rA = rowA < M;
        bool cB = colB < N;
        int  rc = rA ? rowA : (M - 1);
        int  cc = cB ? colB : (N - 1);
        for (int k = kstart; k < K; k += 4) {
            int ka = k + 2 * half;
            int k0 = (ka     < K) ? ka     : K - 1;
            int k1 = (ka + 1 < K) ? ka + 1 : K - 1;
            float a0 = A[(size_t)rc * lda + k0];
            float a1 = A[(size_t)rc * lda + k1];
            float b0 = W[(size_t)k0 * ldw + cc];
            float b1 = W[(size_t)k1 * ldw + cc];
            v2f a, b;
            a.x = (rA && ka     < K) ? a0 : 0.0f;
            a.y = (rA && ka + 1 < K) ? a1 : 0.0f;
            b.x = (cB && ka     < K) ? b0 : 0.0f;
            b.y = (cB && ka + 1 < K) ? b1 : 0.0f;
            acc = __builtin_amdgcn_wmma_f32_16x16x4_f32(false, a, false, b,
                                                        (short)0, acc, false, false);
        }
    }

    int row0 = tm * 16 + 8 * half;
    int col  = tn * 16 + q;
    if (col < N) {
        float bs = bias ? bias[col] : 0.0f;
        for (int r = 0; r < 8; ++r) {
            int row = row0 + r;
            if (row < M) {
                float v = acc[r] + bs;
                if (relu) v = fmaxf(v, 0.0f);
                C[(size_t)row * ldc + col] = v;
            }
        }
    }
}

// ---------------------------------------------------------------------------
// Utility kernels
// ---------------------------------------------------------------------------
__global__ void zero_kernel(float* p, long n)
{
    long i = (long)blockIdx.x * blockDim.x + threadIdx.x;
    if (i < n) p[i] = 0.0f;
}

__global__ void zero_int_kernel(int* p, int n)
{
    int i = blockIdx.x * blockDim.x + threadIdx.x;
    if (i < n) p[i] = 0;
}

__global__ void set_int_kernel(int* p, int v)
{
    if (threadIdx.x == 0 && blockIdx.x == 0) p[0] = v;
}

// Build node features: [emb_atom(64) | deg(16) | charge(16) | hyb(8) | arom(4)
//                       | chiral(4) | (charge scalar, mol only)] padded to 128.
__global__ void embed_kernel(const int* __restrict__ xidx, const float* __restrict__ charge,
                             const float* __restrict__ ea, const float* __restrict__ ed,
                             const float* __restrict__ ec, const float* __restrict__ eh,
                             const float* __restrict__ ear, const float* __restrict__ ech,
                             float* __restrict__ out, int N)
{
    long idx = (long)blockIdx.x * blockDim.x + threadIdx.x;
    if (idx >= (long)N * 128) return;
    int i = (int)(idx >> 7);
    int c = (int)(idx & 127);
    const int* xi = xidx + (size_t)i * 6;
    float v = 0.0f;
    if      (c < 64)  v = ea[xi[0] * 64 + c];
    else if (c < 80)  v = ed[xi[1] * 16 + (c - 64)];
    else if (c < 96)  v = ec[xi[2] * 16 + (c - 80)];
    else if (c < 104) v = eh[xi[3] * 8  + (c - 96)];
    else if (c < 108) v = ear[xi[4] * 4 + (c - 104)];
    else if (c < 112) v = ech[xi[5] * 4 + (c - 108)];
    else if (c == 112 && charge) v = charge[i];
    out[idx] = v;
}

// ---------------------------------------------------------------------------
// GCN degree normalization
// ---------------------------------------------------------------------------
__global__ void deg_init_kernel(float* deg, int N)
{
    int i = blockIdx.x * blockDim.x + threadIdx.x;
    if (i < N) deg[i] = 1.0f;                // self-loop weight
}

__global__ void deg_edge_kernel(const int* __restrict__ ei, const float* __restrict__ ew,
                                float* deg, int E)
{
    int e = blockIdx.x * blockDim.x + threadIdx.x;
    if (e >= E) return;
    int col = ei[(size_t)E + e];
    atomicAdd(&deg[col], ew ? ew[e] : 1.0f);
}

__global__ void deg_invsqrt_kernel(float* deg, int N)
{
    int i = blockIdx.x * blockDim.x + threadIdx.x;
    if (i < N) deg[i] = rsqrtf(deg[i]);      // deg >= 1 always (self loop)
}

// ---------------------------------------------------------------------------
// CSR build for incoming edges (replaces float-atomic scatter with gather)
// ---------------------------------------------------------------------------
__global__ void csr_hist_kernel(const int* __restrict__ ei, int* __restrict__ cnt, int E)
{
    int e = blockIdx.x * blockDim.x + threadIdx.x;
    if (e >= E) return;
    atomicAdd(&cnt[ei[(size_t)E + e]], 1);
}

#define SCAN_B 256
// block-level exclusive scan; per-block totals to bsum
__global__ void scan1_kernel(const int* __restrict__ cnt, int* __restrict__ excl,
                             int* __restrict__ bsum, int N)
{
    __shared__ int sh[SCAN_B];
    int tid = threadIdx.x;
    int gid = blockIdx.x * SCAN_B + tid;
    int v = (gid < N) ? cnt[gid] : 0;
    sh[tid] = v;
    __syncthreads();
    for (int off = 1; off < SCAN_B; off <<= 1) {
        int t = (tid >= off) ? sh[tid - off] : 0;
        __syncthreads();
        sh[tid] += t;
        __syncthreads();
    }
    int incl = sh[tid];
    if (gid < N) excl[gid] = incl - v;
    if (tid == SCAN_B - 1) bsum[blockIdx.x] = incl;
}

// single-block exclusive scan of block sums (nb <= 1024)
__global__ void scan2_kernel(int* __restrict__ bsum, int nb)
{
    __shared__ int sh[1024];
    int tid = threadIdx.x;
    int v = (tid < nb) ? bsum[tid] : 0;
    sh[tid] = v;
    __syncthreads();
    for (int off = 1; off < 1024; off <<= 1) {
        int t = (tid >= off) ? sh[tid - off] : 0;
        __syncthreads();
        sh[tid] += t;
        __syncthreads();
    }
    if (tid < nb) bsum[tid] = sh[tid] - v;
}

__global__ void scan3_kernel(int* __restrict__ excl, const int* __restrict__ bsum, int N)
{
    int gid = blockIdx.x * blockDim.x + threadIdx.x;
    if (gid < N) excl[gid] += bsum[gid >> 8];     // SCAN_B == 256
}

// slot assignment: srcrow/snorm hold source row + fused edge normalization
__global__ void csr_fill_kernel(const int* __restrict__ ei, const float* __restrict__ ew,
                                const float* __restrict__ dis, const int* __restrict__ rowstart,
                                int* __restrict__ fill, int* __restrict__ srcrow,
                                float* __restrict__ snorm, int E)
{
    int e = blockIdx.x * blockDim.x + threadIdx.x;
    if (e >= E) return;
    int r = ei[e];
    int c = ei[(size_t)E + e];
    int slot = rowstart[c] + atomicAdd(&fill[c], 1);
    srcrow[slot] = r;
    snorm[slot]  = dis[r] * (ew ? ew[e] : 1.0f) * dis[c];
}

// out[i] = relu( dis[i]^2*y[i] + sum_j snorm[j]*y[srcrow[j]] + bias )
// one wave per node; lane covers 4 of 128 features (coalesced float4 rows)
__global__ void gcn_gather_kernel(const float* __restrict__ y, const float* __restrict__ dis,
                                  const float* __restrict__ bias,
                                  const int* __restrict__ rowstart,
                                  const int* __restrict__ srcrow,
                                  const float* __restrict__ snorm,
                                  float* __restrict__ out, int N)
{
    long t = (long)blockIdx.x * blockDim.x + threadIdx.x;
    long i = t >> 5;
    int lane = (int)(t & 31);
    if (i >= N) return;
    float d = dis[(int)i];
    float sc = d * d;
    float4 a = ((const float4*)(y + (size_t)i * 128))[lane];
    float4 acc;
    acc.x = a.x * sc; acc.y = a.y * sc; acc.z = a.z * sc; acc.w = a.w * sc;
    int s0 = rowstart[i];
    int s1 = rowstart[i + 1];
    for (int j = s0; j < s1; ++j) {          // wave-uniform trip count
        int r = srcrow[j];
        float nm = snorm[j];
        float4 v = ((const float4*)(y + (size_t)r * 128))[lane];
        acc.x += nm * v.x; acc.y += nm * v.y;
        acc.z += nm * v.z; acc.w += nm * v.w;
    }
    float4 b4 = ((const float4*)bias)[lane];
    acc.x = fmaxf(acc.x + b4.x, 0.0f);
    acc.y = fmaxf(acc.y + b4.y, 0.0f);
    acc.z = fmaxf(acc.z + b4.z, 0.0f);
    acc.w = fmaxf(acc.w + b4.w, 0.0f);
    ((float4*)(out + (size_t)i * 128))[lane] = acc;
}

// ---------------------------------------------------------------------------
// Mean pool over batch segments
// ---------------------------------------------------------------------------
__global__ void pool_accum_kernel(const float* __restrict__ x, const int* __restrict__ bv,
                                  float* __restrict__ sums, float* __restrict__ cnt, int N)
{
    long t = (long)blockIdx.x * blockDim.x + threadIdx.x;
    long i = t >> 5;
    int lane = (int)(t & 31);
    if (i >= N) return;
    int b = bv[i];
    float4 v = ((const float4*)(x + (size_t)i * 128))[lane];
    float* s = sums + (size_t)b * 128 + lane * 4;
    atomicAdd(s + 0, v.x);
    atomicAdd(s + 1, v.y);
    atomicAdd(s + 2, v.z);
    atomicAdd(s + 3, v.w);
    if (lane == 0) atomicAdd(&cnt[b], 1.0f);
}

__global__ void pool_finish_kernel(const float* __restrict__ sums, const float* __restrict__ cnt,
                                   float* __restrict__ feat, int ldf, int B)
{
    int idx = blockIdx.x * blockDim.x + threadIdx.x;
    if (idx >= B * 128) return;
    int b = idx >> 7;
    int c = idx & 127;
    feat[(size_t)b * ldf + c] = sums[idx] / fmaxf(cnt[b], 1.0f);
}

// ---------------------------------------------------------------------------
// 3D conv (SAME, 3x3x3) + BN(batch stats) + ReLU + 2x maxpool, via recompute
// ---------------------------------------------------------------------------
__device__ __forceinline__ float conv3d_eval(const float* __restrict__ in,
                                             const float* __restrict__ w,
                                             const float* __restrict__ cb,
                                             int Cin, int D, int b, int oc,
                                             int z, int y, int x)
{
    float s = cb[oc];
    for (int ic = 0; ic < Cin; ++ic) {
        const float* ip = in + (((size_t)b * Cin + ic) * D * D) * D;
        const float* wp = w + (((size_t)oc * Cin + ic) * 27);
        for (int kz = 0; kz < 3; ++kz) {
            int zz = z + kz - 1; if (zz < 0 || zz >= D) continue;
            for (int ky = 0; ky < 3; ++ky) {
                int yy = y + ky - 1; if (yy < 0 || yy >= D) continue;
                for (int kx = 0; kx < 3; ++kx) {
                    int xx = x + kx - 1; if (xx < 0 || xx >= D) continue;
                    s += ip[((size_t)zz * D + yy) * D + xx] * wp[(kz * 3 + ky) * 3 + kx];
                }
            }
        }
    }
    return s;
}

__global__ void conv_stats_kernel(const float* __restrict__ in, const float* __restrict__ w,
                                  const float* __restrict__ cb, float* __restrict__ csum,
                                  float* __restrict__ csq, int Cin, int Cout, int D, int B)
{
    long DDD = (long)D * D * D;
    long tot = (long)B * Cout * DDD;
    long idx = (long)blockIdx.x * blockDim.x + threadIdx.x;
    float v = 0.0f, v2 = 0.0f;
    int oc = 0;
    if (idx < tot) {
        long s = idx % DDD;
        int x = (int)(s % D);
        int y = (int)((s / D) % D);
        int z = (int)(s / ((long)D * D));
        oc    = (int)((idx / DDD) % Cout);
        int b = (int)(idx / (DDD * Cout));
        v  = conv3d_eval(in, w, cb, Cin, D, b, oc, z, y, x);
        v2 = v * v;
    }
    for (int off = 16; off > 0; off >>= 1) {
        v  += __shfl_down(v,  off, 32);
        v2 += __shfl_down(v2, off, 32);
    }
    if ((threadIdx.x & 31) == 0 && idx < tot) {
        atomicAdd(&csum[oc], v);
        atomicAdd(&csq[oc],  v2);
    }
}

__global__ void bn_finalize_kernel(const float* __restrict__ csum, const float* __restrict__ csq,
                                   const float* __restrict__ g, const float* __restrict__ bb,
                                   float* __restrict__ scl, float* __restrict__ shf,
                                   int Cout, float inv_count)
{
    int c = blockIdx.x * blockDim.x + threadIdx.x;
    if (c >= Cout) return;
    float m   = csum[c] * inv_count;
    float var = csq[c] * inv_count - m * m;
    float s   = g[c] * rsqrtf(var + 1e-5f);
    scl[c] = s;
    shf[c] = bb[c] - m * s;
}

__global__ void conv_pool_kernel(const float* __restrict__ in, const float* __restrict__ w,
                                 const float* __restrict__ cb, const float* __restrict__ scl,
                                 const float* __restrict__ shf, float* __restrict__ out,
                                 int Cin, int Cout, int D, int B)
{
    int Do = D >> 1;
    long DDD = (long)Do * Do * Do;
    long tot = (long)B * Cout * DDD;
    long idx = (long)blockIdx.x * blockDim.x + threadIdx.x;
    if (idx >= tot) return;
    long s = idx % DDD;
    int x2 = (int)(s % Do) * 2;
    int y2 = (int)((s / Do) % Do) * 2;
    int z2 = (int)(s / ((long)Do * Do)) * 2;
    int oc = (int)((idx / DDD) % Cout);
    int b  = (int)(idx / (DDD * Cout));
    float sc = scl[oc], sh = shf[oc];
    float mx = 0.0f;
    for (int dz = 0; dz < 2; ++dz)
        for (int dy = 0; dy < 2; ++dy)
            for (int dx = 0; dx < 2; ++dx) {
                float v = conv3d_eval(in, w, cb, Cin, D, b, oc, z2 + dz, y2 + dy, x2 + dx);
                mx = fmaxf(mx, fmaxf(v * sc + sh, 0.0f));
            }
    out[idx] = mx;
}

__global__ void bn1d_relu_kernel(const float* __restrict__ x, const float* __restrict__ g,
                                 const float* __restrict__ bb, float* __restrict__ out,
                                 int ldo, int B, int C)
{
    int c = blockIdx.x * blockDim.x + threadIdx.x;
    if (c >= C) return;
    float m = 0.0f;
    for (int b = 0; b < B; ++b) m += x[(size_t)b * C + c];
    m /= (float)B;
    float v = 0.0f;
    for (int b = 0; b < B; ++b) {
        float d = x[(size_t)b * C + c] - m;
        v += d * d;
    }
    v /= (float)B;
    float s = g[c] * rsqrtf(v + 1e-5f);
    for (int b = 0; b < B; ++b)
        out[(size_t)b * ldo + c] = fmaxf((x[(size_t)b * C + c] - m) * s + bb[c], 0.0f);
}

// ---------------------------------------------------------------------------
// Orchestration
// ---------------------------------------------------------------------------
extern "C" void kernel_launch(void* const* d_in, const int* in_sizes, int n_in,
                              void* d_out, int out_size, void* d_ws, size_t ws_size,
                              hipStream_t stream)
{
    const int*   mol_x  = (const int*)  d_in[0];
    const float* mol_q  = (const float*)d_in[1];
    const int*   mol_ei = (const int*)  d_in[2];
    const float* mol_ew = (const float*)d_in[3];
    const int*   mol_bv = (const int*)  d_in[4];
    const int*   zeo_x  = (const int*)  d_in[5];
    const int*   zeo_ei = (const int*)  d_in[6];
    const int*   zeo_bv = (const int*)  d_in[7];
    const float* voxel  = (const float*)d_in[8];
    const float* gattr  = (const float*)d_in[9];
    const float* ea  = (const float*)d_in[10];
    const float* ed  = (const float*)d_in[11];
    const float* ec  = (const float*)d_in[12];
    const float* eh  = (const float*)d_in[13];
    const float* ear = (const float*)d_in[14];
    const float* ech = (const float*)d_in[15];
    const float* mc1_w = (const float*)d_in[16]; const float* mc1_b = (const float*)d_in[17];
    const float* mc2_w = (const float*)d_in[18]; const float* mc2_b = (const float*)d_in[19];
    const float* zc1_w = (const float*)d_in[20]; const float* zc1_b = (const float*)d_in[21];
    const float* zc2_w = (const float*)d_in[22]; const float* zc2_b = (const float*)d_in[23];
    const float* cv1_w = (const float*)d_in[24]; const float* cv1_b = (const float*)d_in[25];
    const float* bn1_g = (const float*)d_in[26]; const float* bn1_b = (const float*)d_in[27];
    const float* cv2_w = (const float*)d_in[28]; const float* cv2_b = (const float*)d_in[29];
    const float* bn2_g = (const float*)d_in[30]; const float* bn2_b = (const float*)d_in[31];
    const float* cv3_w = (const float*)d_in[32]; const float* cv3_b = (const float*)d_in[33];
    const float* bn3_g = (const float*)d_in[34]; const float* bn3_b = (const float*)d_in[35];
    const float* fc_w  = (const float*)d_in[36]; const float* fc_b  = (const float*)d_in[37];
    const float* ge1_w = (const float*)d_in[38]; const float* ge1_b = (const float*)d_in[39];
    const float* ge2_w = (const float*)d_in[40]; const float* ge2_b = (const float*)d_in[41];
    const float* gebn_g = (const float*)d_in[42]; const float* gebn_b = (const float*)d_in[43];
    const float* h1_w = (const float*)d_in[44]; const float* h1_b = (const float*)d_in[45];
    const float* h2_w = (const float*)d_in[46]; const float* h2_b = (const float*)d_in[47];
    const float* h3_w = (const float*)d_in[48]; const float* h3_b = (const float*)d_in[49];

    const int Nm = in_sizes[0] / 6, Em = in_sizes[2] / 2;
    const int Nz = in_sizes[5] / 6, Ez = in_sizes[6] / 2;
    const int Bn = in_sizes[9] / 17;
    const int Nmax = (Nm > Nz) ? Nm : Nz;
    const int Emax = (Em > Ez) ? Em : Ez;

    float* ws = (float*)d_ws;
    const long NM128 = (long)Nmax * 128;
    float* bufA = ws;
    float* bufB = ws + NM128;
    float* dis  = bufB + NM128;
    float* sums = dis + Nmax;
    float* cnt  = sums + (long)Bn * 128;
    float* csum = cnt + Bn;
    float* csq  = csum + 64;
    float* scl  = csq + 64;
    float* shf  = scl + 64;
    float* FEAT = shf + 64;               // [B,512]: mol|zeo|glob|voxel
    float* G1   = FEAT + (long)Bn * 512;  // [B,64]
    float* G2   = G1 + (long)Bn * 64;     // [B,128]
    float* H1   = G2 + (long)Bn * 128;    // [B,512]
    float* H2   = H1 + (long)Bn * 512;    // [B,256]
    // CSR region (ints/floats)
    int*   rowstart = (int*)(H2 + (long)Bn * 256);   // [Nmax+1]
    int*   fillc    = rowstart + (Nmax + 1);         // [Nmax]
    int*   srcrow   = fillc + Nmax;                  // [Emax]
    float* snorm    = (float*)(srcrow + Emax);       // [Emax]
    int*   bsum     = (int*)(snorm + Emax);          // [<=1024]
    // CNN pooled maps reuse the big ping-pong buffers
    float* P1   = bufA;                               // [B,16,32^3]
    float* P2   = bufB;                               // [B,32,16^3]
    float* P3   = bufB + (long)Bn * 32 * 16 * 16 * 16; // [B,64,8^3]

    auto nblk = [](long n) { return (unsigned)((n + TPB - 1) / TPB); };
    auto zero = [&](float* p, long n) {
        zero_kernel<<<nblk(n), TPB, 0, stream>>>(p, n);
    };
    auto gemm = [&](const float* A, int lda, const float* W, int ldw, const float* bias,
                    float* C, int ldc, int M, int N, int K, int relu) {
        int tiles = ((M + 15) / 16) * ((N + 15) / 16);
        wmma_gemm_kernel<<<(tiles + 7) / 8, TPB, 0, stream>>>(A, lda, W, ldw, bias,
                                                              C, ldc, M, N, K, relu);
    };

    // ---- one GCN layer: bufA (x) -> gemm -> gather+bias+relu -> bufA ----
    auto gcn_layer = [&](const float* W, const float* bias, int N, int E, int K) {
        gemm(bufA, 128, W, 128, nullptr, bufB, 128, N, 128, K, 0);           // y = x@W (WMMA)
        gcn_gather_kernel<<<nblk((long)N * 32), TPB, 0, stream>>>(
            bufB, dis, bias, rowstart, srcrow, snorm, bufA, N);
    };

    // ---- full GNN branch -> FEAT[:, off:off+128] ----
    auto branch = [&](const int* xidx, const float* charge, const int* ei, const float* ew,
                      const int* bv, const float* W1, const float* b1,
                      const float* W2, const float* b2, int N, int E, int K1, float* featdst) {
        embed_kernel<<<nblk((long)N * 128), TPB, 0, stream>>>(xidx, charge, ea, ed, ec,
                                                              eh, ear, ech, bufA, N);
        // symmetric degree normalization (incl. self loops)
        deg_init_kernel<<<nblk(N), TPB, 0, stream>>>(dis, N);
        deg_edge_kernel<<<nblk(E), TPB, 0, stream>>>(ei, ew, dis, E);
        deg_invsqrt_kernel<<<nblk(N), TPB, 0, stream>>>(dis, N);
        // CSR of incoming edges (reused by both layers)
        int nb = (N + SCAN_B - 1) / SCAN_B;
        zero_int_kernel<<<nblk(N), TPB, 0, stream>>>(fillc, N);
        csr_hist_kernel<<<nblk(E), TPB, 0, stream>>>(ei, fillc, E);
        scan1_kernel<<<nb, SCAN_B, 0, stream>>>(fillc, rowstart, bsum, N);
        scan2_kernel<<<1, 1024, 0, stream>>>(bsum, nb);
        scan3_kernel<<<nblk(N), TPB, 0, stream>>>(rowstart, bsum, N);
        set_int_kernel<<<1, 1, 0, stream>>>(rowstart + N, E);
        zero_int_kernel<<<nblk(N), TPB, 0, stream>>>(fillc, N);
        csr_fill_kernel<<<nblk(E), TPB, 0, stream>>>(ei, ew, dis, rowstart,
                                                     fillc, srcrow, snorm, E);
        // two GCN layers (WMMA gemm + gather)
        gcn_layer(W1, b1, N, E, K1);
        gcn_layer(W2, b2, N, E, 128);
        // mean pool
        zero(sums, (long)Bn * 128);
        zero(cnt, Bn);
        pool_accum_kernel<<<nblk((long)N * 32), TPB, 0, stream>>>(bufA, bv, sums, cnt, N);
        pool_finish_kernel<<<nblk((long)Bn * 128), TPB, 0, stream>>>(sums, cnt, featdst, 512, Bn);
    };

    // ---- conv + BN + relu + pool layer (two-pass, recompute) ----
    auto conv_layer = [&](const float* in, const float* w, const float* cb,
                          const float* g, const float* bb, int Cin, int Cout, int D, float* out) {
        zero(csum, Cout);
        zero(csq, Cout);
        long tot = (long)Bn * Cout * D * D * D;
        conv_stats_kernel<<<nblk(tot), TPB, 0, stream>>>(in, w, cb, csum, csq, Cin, Cout, D, Bn);
        bn_finalize_kernel<<<1, 64, 0, stream>>>(csum, csq, g, bb, scl, shf, Cout,
                                                 1.0f / ((float)Bn * D * D * D));
        long tp = (long)Bn * Cout * (D / 2) * (D / 2) * (D / 2);
        conv_pool_kernel<<<nblk(tp), TPB, 0, stream>>>(in, w, cb, scl, shf, out, Cin, Cout, D, Bn);
    };

    // ===== molecule + zeolite GNN branches =====
    branch(mol_x, mol_q, mol_ei, mol_ew, mol_bv, mc1_w, mc1_b, mc2_w, mc2_b,
           Nm, Em, 113, FEAT + 0);
    branch(zeo_x, nullptr, zeo_ei, nullptr, zeo_bv, zc1_w, zc1_b, zc2_w, zc2_b,
           Nz, Ez, 112, FEAT + 128);

    // ===== voxel 3D CNN (bufA/bufB now free) =====
    conv_layer(voxel, cv1_w, cv1_b, bn1_g, bn1_b, 2, 16, 64, P1);
    conv_layer(P1, cv2_w, cv2_b, bn2_g, bn2_b, 16, 32, 32, P2);
    conv_layer(P2, cv3_w, cv3_b, bn3_g, bn3_b, 32, 64, 16, P3);
    gemm(P3, 32768, fc_w, 128, fc_b, FEAT + 384, 512, Bn, 128, 32768, 1);   // feat_v

    // ===== global encoder =====
    gemm(gattr, 17, ge1_w, 64, ge1_b, G1, 64, Bn, 64, 17, 1);
    gemm(G1, 64, ge2_w, 128, ge2_b, G2, 128, Bn, 128, 64, 0);
    bn1d_relu_kernel<<<1, 128, 0, stream>>>(G2, gebn_g, gebn_b, FEAT + 256, 512, Bn, 128);

    // ===== fusion head =====
    gemm(FEAT, 512, h1_w, 512, h1_b, H1, 512, Bn, 512, 512, 1);
    gemm(H1, 512, h2_w, 256, h2_b, H2, 256, Bn, 256, 512, 1);
    gemm(H2, 256, h3_w, 3, h3_b, (float*)d_out, 3, Bn, 3, 256, 0);
}